// UltraSparseExpertLayer_64630667870636
// MI455X (gfx1250) — compile-verified
//
#include <hip/hip_runtime.h>
#include <math.h>

// ---- Problem constants (from reference) ----
#define D_MODEL   512
#define D_EXPERT  1024
#define N_EXPERTS 64
#define TOP_K     2
#define T_TOK     512          // BATCH*SEQ
#define OUT_ELEMS (T_TOK * D_MODEL)   // 262144, aux at index OUT_ELEMS

typedef __attribute__((ext_vector_type(2))) float v2f;
typedef __attribute__((ext_vector_type(8))) float v8f;

// ---------------------------------------------------------------------------
// K0: zero output + expert counts
// ---------------------------------------------------------------------------
__global__ void k_init(float* __restrict__ out, int* __restrict__ counts) {
    int i = blockIdx.x * blockDim.x + threadIdx.x;
    if (i < OUT_ELEMS + 1) out[i] = 0.0f;
    if (i < N_EXPERTS) counts[i] = 0;
}

// ---------------------------------------------------------------------------
// K1: router. One block (64 threads) per token. Thread e computes logit_e.
// ---------------------------------------------------------------------------
__global__ __launch_bounds__(N_EXPERTS) void k_router(
    const float* __restrict__ x, const float* __restrict__ rw,
    float* __restrict__ probs, int* __restrict__ top_idx,
    float* __restrict__ top_w, int* __restrict__ counts) {
    int t = blockIdx.x;
    int e = threadIdx.x;

    const float4* xr = (const float4*)(x + (size_t)t * D_MODEL);
    const float4* wr = (const float4*)(rw + (size_t)e * D_MODEL);
    float acc = 0.0f;
#pragma unroll 4
    for (int i = 0; i < D_MODEL / 4; ++i) {
        float4 a = xr[i], b = wr[i];
        acc += a.x * b.x + a.y * b.y + a.z * b.z + a.w * b.w;
    }

    __shared__ float lg[N_EXPERTS];
    __shared__ float ex[N_EXPERTS];
    __shared__ float s_max, s_sum;
    __shared__ int   s_i1, s_i2;
    __shared__ float s_v1, s_v2;

    lg[e] = acc;
    __syncthreads();

    if (e == 0) {
        // deterministic sequential top-2 (first occurrence on ties, like lax.top_k)
        float v1 = -3.4e38f; int i1 = 0;
        for (int i = 0; i < N_EXPERTS; ++i) if (lg[i] > v1) { v1 = lg[i]; i1 = i; }
        float v2 = -3.4e38f; int i2 = (i1 == 0) ? 1 : 0;
        for (int i = 0; i < N_EXPERTS; ++i) if (i != i1 && lg[i] > v2) { v2 = lg[i]; i2 = i; }
        s_v1 = v1; s_v2 = v2; s_i1 = i1; s_i2 = i2; s_max = v1;
    }
    __syncthreads();

    ex[e] = expf(lg[e] - s_max);
    __syncthreads();
    if (e == 0) {
        float s = 0.0f;
        for (int i = 0; i < N_EXPERTS; ++i) s += ex[i];   // deterministic order
        s_sum = s;
    }
    __syncthreads();
    probs[(size_t)t * N_EXPERTS + e] = ex[e] / s_sum;

    if (e == 0) {
        float e2 = expf(s_v2 - s_v1);       // softmax over the two selected logits
        float s  = 1.0f + e2;
        top_idx[t * 2 + 0] = s_i1;  top_w[t * 2 + 0] = 1.0f / s;
        top_idx[t * 2 + 1] = s_i2;  top_w[t * 2 + 1] = e2 / s;
        atomicAdd(&counts[s_i1], 1);
        atomicAdd(&counts[s_i2], 1);
    }
}

// ---------------------------------------------------------------------------
// K2: exclusive scan of counts -> offsets[65]; zero fill counters. 1 thread.
// ---------------------------------------------------------------------------
__global__ void k_scan(const int* __restrict__ counts, int* __restrict__ offsets,
                       int* __restrict__ fill) {
    if (threadIdx.x == 0) {
        int off = 0;
        for (int e = 0; e < N_EXPERTS; ++e) {
            offsets[e] = off;
            off += counts[e];
            fill[e] = 0;
        }
        offsets[N_EXPERTS] = off;   // == T_TOK*TOP_K == 1024
    }
}

// ---------------------------------------------------------------------------
// K3: scatter tokens into compact per-expert row lists.
// ---------------------------------------------------------------------------
__global__ void k_fill(const int* __restrict__ top_idx, const float* __restrict__ top_w,
                       const int* __restrict__ offsets, int* __restrict__ fill,
                       int* __restrict__ tok_of_row, float* __restrict__ wt_of_row) {
    int t = blockIdx.x * blockDim.x + threadIdx.x;
    if (t >= T_TOK) return;
#pragma unroll
    for (int k = 0; k < TOP_K; ++k) {
        int e = top_idx[t * 2 + k];
        int pos = atomicAdd(&fill[e], 1);
        int row = offsets[e] + pos;
        tok_of_row[row] = t;
        wt_of_row[row]  = top_w[t * 2 + k];
    }
}

// ---------------------------------------------------------------------------
// K4: gate/up GEMMs + fused SiLU -> h_buf[row][F].
// grid: x = f_tile (16 of 64F), y = token_tile (16 of 32 rows), z = expert.
// 128 threads = 4 waves; wave owns 16 F cols, 32 token rows (2 WMMA M-halves).
// One B fragment (gate + up) feeds both M halves -> weights stream once per
// expert (cnt<=32 in the typical case) and 4 independent WMMA chains/wave.
// ---------------------------------------------------------------------------
__global__ __launch_bounds__(128) void k_gateup(
    const float* __restrict__ x, const float* __restrict__ wg,
    const float* __restrict__ wu, const int* __restrict__ offsets,
    const int* __restrict__ tok_of_row, float* __restrict__ h_buf) {
    const int e  = blockIdx.z;
    const int tt = blockIdx.y;
    const int ft = blockIdx.x;
    const int base = offsets[e];
    const int cnt  = offsets[e + 1] - base;
    if (tt * 32 >= cnt) return;                    // block-uniform exit
    const int nv = min(32, cnt - tt * 32);

    const int tid = threadIdx.x;
    __shared__ float xs[32 * 516];                 // stride 516: conflict-free rows
    __shared__ int   toks[32];
    if (tid < 32) toks[tid] = (tid < nv) ? tok_of_row[base + tt * 32 + tid] : -1;
    __syncthreads();
    // 128 threads x float4 == one 512-float row per iteration
    for (int it = 0; it < 32; ++it) {
        int tk = toks[it];
        float4 v = make_float4(0.f, 0.f, 0.f, 0.f);
        if (tk >= 0) v = *(const float4*)(x + (size_t)tk * D_MODEL + tid * 4);
        *(float4*)(&xs[it * 516 + tid * 4]) = v;
    }
    __syncthreads();

    const int w  = tid >> 5;
    const int l  = tid & 31;
    const int hi = l >> 4;       // K-half select (K: 2*hi, 2*hi+1)
    const int mn = l & 15;       // A row / B column within tile
    const int fcol = ft * 64 + w * 16 + mn;

    const float* arow0 = &xs[mn * 516 + 2 * hi];           // M rows 0..15
    const float* arow1 = &xs[(mn + 16) * 516 + 2 * hi];    // M rows 16..31
    const float* bg = wg + ((size_t)e * D_EXPERT + fcol) * D_MODEL + 2 * hi;
    const float* bu = wu + ((size_t)e * D_EXPERT + fcol) * D_MODEL + 2 * hi;

    v8f accg0 = {}, accg1 = {};
    v8f accu0 = {}, accu1 = {};
#pragma unroll 4
    for (int k0 = 0; k0 < D_MODEL; k0 += 4) {
        v2f a0 = *(const v2f*)(arow0 + k0);
        v2f a1 = *(const v2f*)(arow1 + k0);
        v2f g  = *(const v2f*)(bg + k0);
        v2f u  = *(const v2f*)(bu + k0);
        accg0 = __builtin_amdgcn_wmma_f32_16x16x4_f32(false, a0, false, g,
                                                      (short)0, accg0, false, false);
        accg1 = __builtin_amdgcn_wmma_f32_16x16x4_f32(false, a1, false, g,
                                                      (short)0, accg1, false, false);
        accu0 = __builtin_amdgcn_wmma_f32_16x16x4_f32(false, a0, false, u,
                                                      (short)0, accu0, false, false);
        accu1 = __builtin_amdgcn_wmma_f32_16x16x4_f32(false, a1, false, u,
                                                      (short)0, accu1, false, false);
    }

#pragma unroll
    for (int r = 0; r < 8; ++r) {
        int m = r + 8 * hi;                         // C layout: vgpr r, lane-half hi
        if (m < nv) {
            float g = accg0[r];
            float h = (g / (1.0f + expf(-g))) * accu0[r];   // silu(g) * up
            h_buf[(size_t)(base + tt * 32 + m) * D_EXPERT + fcol] = h;
        }
        int m2 = m + 16;
        if (m2 < nv) {
            float g = accg1[r];
            float h = (g / (1.0f + expf(-g))) * accu1[r];
            h_buf[(size_t)(base + tt * 32 + m2) * D_EXPERT + fcol] = h;
        }
    }
}

// ---------------------------------------------------------------------------
// K5: down projection + weighted scatter into out (2 commutative f32 atomics
// per out element -> deterministic).
// grid: x = d_tile (8 of 64D), y = token_tile (16 of 32 rows), z = expert.
// 4 independent WMMA chains: 2 M-halves x even/odd K groups.
// ---------------------------------------------------------------------------
__global__ __launch_bounds__(128) void k_down(
    const float* __restrict__ h_buf, const float* __restrict__ wd,
    const int* __restrict__ offsets, const int* __restrict__ tok_of_row,
    const float* __restrict__ wt_of_row, float* __restrict__ out) {
    const int e  = blockIdx.z;
    const int tt = blockIdx.y;
    const int dt = blockIdx.x;
    const int base = offsets[e];
    const int cnt  = offsets[e + 1] - base;
    if (tt * 32 >= cnt) return;                    // block-uniform exit
    const int nv = min(32, cnt - tt * 32);

    const int tid = threadIdx.x;
    const int w  = tid >> 5;
    const int l  = tid & 31;
    const int hi = l >> 4;
    const int mn = l & 15;
    const int dcol = dt * 64 + w * 16 + mn;

    const float* arow0 = h_buf + (size_t)(base + tt * 32 + mn) * D_EXPERT + 2 * hi;
    const float* arow1 = arow0 + (size_t)16 * D_EXPERT;
    const float* brow  = wd + ((size_t)e * D_MODEL + dcol) * D_EXPERT + 2 * hi;

    v8f acc00 = {}, acc01 = {};   // M rows 0..15, even/odd K groups
    v8f acc10 = {}, acc11 = {};   // M rows 16..31
#pragma unroll 2
    for (int f0 = 0; f0 < D_EXPERT; f0 += 8) {
        v2f a00 = *(const v2f*)(arow0 + f0);
        v2f a01 = *(const v2f*)(arow0 + f0 + 4);
        v2f a10 = *(const v2f*)(arow1 + f0);
        v2f a11 = *(const v2f*)(arow1 + f0 + 4);
        v2f b0  = *(const v2f*)(brow + f0);
        v2f b1  = *(const v2f*)(brow + f0 + 4);
        acc00 = __builtin_amdgcn_wmma_f32_16x16x4_f32(false, a00, false, b0,
                                                      (short)0, acc00, false, false);
        acc10 = __builtin_amdgcn_wmma_f32_16x16x4_f32(false, a10, false, b0,
                                                      (short)0, acc10, false, false);
        acc01 = __builtin_amdgcn_wmma_f32_16x16x4_f32(false, a01, false, b1,
                                                      (short)0, acc01, false, false);
        acc11 = __builtin_amdgcn_wmma_f32_16x16x4_f32(false, a11, false, b1,
                                                      (short)0, acc11, false, false);
    }

#pragma unroll
    for (int r = 0; r < 8; ++r) {
        int m = r + 8 * hi;
        if (m < nv) {
            int row = base + tt * 32 + m;
            unsafeAtomicAdd(&out[(size_t)tok_of_row[row] * D_MODEL + dcol],
                            wt_of_row[row] * (acc00[r] + acc01[r]));
        }
        int m2 = m + 16;
        if (m2 < nv) {
            int row = base + tt * 32 + m2;
            unsafeAtomicAdd(&out[(size_t)tok_of_row[row] * D_MODEL + dcol],
                            wt_of_row[row] * (acc10[r] + acc11[r]));
        }
    }
}

// ---------------------------------------------------------------------------
// K6: aux load-balancing loss. Deterministic sequential reductions.
// ---------------------------------------------------------------------------
__global__ __launch_bounds__(N_EXPERTS) void k_aux(const float* __restrict__ probs,
                                                   float* __restrict__ out_aux) {
    __shared__ float pe[N_EXPERTS];
    int e = threadIdx.x;
    float s = 0.0f;
    for (int t = 0; t < T_TOK; ++t) s += probs[(size_t)t * N_EXPERTS + e];
    pe[e] = (s / (float)T_TOK) * (float)N_EXPERTS;
    __syncthreads();
    if (e == 0) {
        float m = 0.0f;
        for (int i = 0; i < N_EXPERTS; ++i) m += pe[i];
        m /= (float)N_EXPERTS;
        float v = 0.0f;
        for (int i = 0; i < N_EXPERTS; ++i) { float d = pe[i] - m; v += d * d; }
        out_aux[0] = v / (float)(N_EXPERTS - 1);   // unbiased (ddof=1)
    }
}

// ---------------------------------------------------------------------------
extern "C" void kernel_launch(void* const* d_in, const int* in_sizes, int n_in,
                              void* d_out, int out_size, void* d_ws, size_t ws_size,
                              hipStream_t stream) {
    (void)in_sizes; (void)n_in; (void)out_size; (void)ws_size;
    const float* x        = (const float*)d_in[0];   // [1,512,512]
    const float* router_w = (const float*)d_in[1];   // [64,512]
    const float* w_gate   = (const float*)d_in[2];   // [64,1024,512]
    const float* w_up     = (const float*)d_in[3];   // [64,1024,512]
    const float* w_down   = (const float*)d_in[4];   // [64,512,1024]
    float* out = (float*)d_out;                      // 262144 + 1 (aux)

    // ---- workspace layout ----
    char* p = (char*)d_ws;
    float* probs      = (float*)p; p += (size_t)T_TOK * N_EXPERTS * 4;   // 128 KB
    int*   top_idx    = (int*)p;   p += (size_t)T_TOK * TOP_K * 4;
    float* top_w      = (float*)p; p += (size_t)T_TOK * TOP_K * 4;
    int*   counts     = (int*)p;   p += N_EXPERTS * 4;
    int*   offsets    = (int*)p;   p += (N_EXPERTS + 1) * 4;
    int*   fill       = (int*)p;   p += N_EXPERTS * 4;
    int*   tok_of_row = (int*)p;   p += (size_t)T_TOK * TOP_K * 4;
    float* wt_of_row  = (float*)p; p += (size_t)T_TOK * TOP_K * 4;
    p = (char*)(((uintptr_t)p + 255) & ~(uintptr_t)255);
    float* h_buf      = (float*)p; p += (size_t)1056 * D_EXPERT * 4;     // slack for padded-row reads

    k_init<<<(OUT_ELEMS + 1 + 255) / 256, 256, 0, stream>>>(out, counts);
    k_router<<<T_TOK, N_EXPERTS, 0, stream>>>(x, router_w, probs, top_idx, top_w, counts);
    k_scan<<<1, 32, 0, stream>>>(counts, offsets, fill);
    k_fill<<<(T_TOK + 255) / 256, 256, 0, stream>>>(top_idx, top_w, offsets, fill,
                                                    tok_of_row, wt_of_row);
    // f_tiles=16, token_tiles=16 (32 rows each, max 512 rows/expert), experts=64
    k_gateup<<<dim3(16, 16, 64), 128, 0, stream>>>(x, w_gate, w_up, offsets,
                                                   tok_of_row, h_buf);
    // d_tiles=8
    k_down<<<dim3(8, 16, 64), 128, 0, stream>>>(h_buf, w_down, offsets,
                                                tok_of_row, wt_of_row, out);
    k_aux<<<1, N_EXPERTS, 0, stream>>>(probs, out + OUT_ELEMS);
}